// FinalizedQuantizedLinear_68599217651720
// MI455X (gfx1250) — compile-verified
//
#include <hip/hip_runtime.h>
#include <hip/hip_bf16.h>

typedef __attribute__((ext_vector_type(2))) float v2f;
typedef __attribute__((ext_vector_type(8))) float v8f;
typedef __attribute__((ext_vector_type(4))) int   v4i;

#define O_FEATS   11008
#define IN_FEATS  4096
#define NGROUPS   512          // IN_FEATS / 8
#define TILE_O    16
#define NWAVES    4
#define G_PER_WAVE (NGROUPS / NWAVES)   // 128

// One block = 4 waves (wave32). Block handles a 16-wide slice of out-features.
// Each wave accumulates a 16x16 fp32 tile (M=16 padded from 4, N=16 out feats)
// over its 128 in-groups using V_WMMA_F32_16X16X4_F32 (2 WMMAs per group, K=8).
// Codes are consumed 4 groups at a time via one b128 load per lane.
__global__ __launch_bounds__(128)
void aqlm_fused_gemv_wmma(const float* __restrict__ x,
                          const float* __restrict__ cb,      // [65536][8]
                          const int*   __restrict__ codes,   // [11008][512]
                          const float* __restrict__ scales,  // [11008]
                          const float* __restrict__ bias,    // [11008]
                          float*       __restrict__ out) {   // [4][11008]
  const int tid  = (int)threadIdx.x;
  const int lane = tid & 31;
  const int wid  = tid >> 5;
  const int o0   = (int)blockIdx.x * TILE_O;

  const int   n    = lane & 15;          // out-feature within tile / A-row M
  const int   half = lane >> 4;          // 0: K={0,1}, 1: K={2,3} of each 4-slab
  const int   kp   = half << 1;
  const int   arow = n & 3;              // clamped x row for safe loads
  const float amask = (n < 4) ? 1.0f : 0.0f;   // zero A rows 4..15

  const float* xrow = x + arow * IN_FEATS;
  const int*   crow = codes + (o0 + n) * NGROUPS;

  const int gbeg = wid * G_PER_WAVE;
  const int gend = gbeg + G_PER_WAVE;

  // Pull this wave's whole code window (128 groups x 4B per row = 4 lines/row)
  // toward the caches before entering the dependent-gather loop.
#pragma unroll
  for (int p = 0; p < 4; ++p)
    __builtin_prefetch(crow + gbeg + 32 * p, 0, 0);

  v8f acc = {};

  for (int g = gbeg; g < gend; g += 4) {
    const v4i c4 = *(const v4i*)(crow + g);   // global_load_b128: 4 codes
#pragma unroll
    for (int u = 0; u < 4; ++u) {
      const v2f* e2 = (const v2f*)(cb + (c4[u] << 3)); // 32B entry, L2-resident
      v2f b0 = e2[half];                                // w[kp..kp+1]   (K-slab 0)
      v2f b1 = e2[half + 2];                            // w[kp+4..kp+5] (K-slab 1)

      const int kbase = ((g + u) << 3) + kp;
      v2f a0 = *(const v2f*)(xrow + kbase);
      v2f a1 = *(const v2f*)(xrow + kbase + 4);
      a0 *= amask;
      a1 *= amask;

      // 8 args: (neg_a, A, neg_b, B, c_mod, C, reuse_a, reuse_b)
      acc = __builtin_amdgcn_wmma_f32_16x16x4_f32(false, a0, false, b0,
                                                  (short)0, acc, false, false);
      acc = __builtin_amdgcn_wmma_f32_16x16x4_f32(false, a1, false, b1,
                                                  (short)0, acc, false, false);
    }
  }

  // Cross-wave K reduction through LDS (stride-9 padding: conflict-free).
  __shared__ float red[NWAVES][32][9];
#pragma unroll
  for (int j = 0; j < 8; ++j) red[wid][lane][j] = acc[j];
  __syncthreads();

  // C/D layout: lane L (0..15), VGPR r in 0..3  ->  C[M=r][N=L].
  if (tid < 16) {
    const float s = scales[o0 + tid];
    const float b = bias[o0 + tid];
#pragma unroll
    for (int r = 0; r < 4; ++r) {
      float v = red[0][tid][r] + red[1][tid][r] + red[2][tid][r] + red[3][tid][r];
      out[r * O_FEATS + o0 + tid] = s * v + b;
    }
  }
}

extern "C" void kernel_launch(void* const* d_in, const int* in_sizes, int n_in,
                              void* d_out, int out_size, void* d_ws, size_t ws_size,
                              hipStream_t stream) {
  const float* x      = (const float*)d_in[0];
  const float* cbks   = (const float*)d_in[1];
  const int*   codes  = (const int*)d_in[2];
  const float* scales = (const float*)d_in[3];
  const float* bias   = (const float*)d_in[4];
  float*       out    = (float*)d_out;

  dim3 grid(O_FEATS / TILE_O);   // 688 blocks
  dim3 block(128);               // 4 wave32 waves
  hipLaunchKernelGGL(aqlm_fused_gemv_wmma, grid, block, 0, stream,
                     x, cbks, codes, scales, bias, out);
}